// CausalHaarContext_34359738368291
// MI455X (gfx1250) — compile-verified
//
#include <hip/hip_runtime.h>

// CausalHaarContext for MI455X (gfx1250, wave32).
// 3-phase parallel-scan chunked causal linear attention, all matmuls via
// V_WMMA_F32_16X16X4_F32 (fp32 preserved; problem is HBM-bound anyway).

typedef float v2f __attribute__((ext_vector_type(2)));
typedef float v8f __attribute__((ext_vector_type(8)));

#define BQ 2
#define HQ 16
#define NQ 4096
#define DQ 64
#define LQ 4
#define CQ 64                      // internal chunk (math-equivalent to any chunking)
#define NCQ (NQ / CQ)              // 64 chunks
#define BHQ (BQ * HQ)              // 32
#define PS (DQ + 4)                // padded LDS row stride (bank-conflict avoidance)
#define STATE_STRIDE (DQ * DQ + DQ) // 4160 floats per (bh,l,chunk) state
#define EPSQ 1e-6f

__device__ __forceinline__ float elu1(float x) {
    return x > 0.0f ? x + 1.0f : __expf(x);
}

// One K=4 step of a 16x16 f32 WMMA tile.
__device__ __forceinline__ v8f wmma4(v8f acc, float ax, float ay, float bx, float by) {
    v2f a; a.x = ax; a.y = ay;
    v2f b; b.x = bx; b.y = by;
    return __builtin_amdgcn_wmma_f32_16x16x4_f32(
        /*neg_a=*/false, a, /*neg_b=*/false, b,
        /*c_mod=*/(short)0, acc, /*reuse_a=*/false, /*reuse_b=*/false);
}

// ---------------------------------------------------------------------------
// Kernel 1: per-(bh, chunk) per-level state contributions:
//   KV[d][e] = sum_c kp[c][d]*vp[c][e],  Ksum[d] = sum_c kp[c][d]
// where kp = elu(blockmean(k) @ Wk^T)+1, vp = blockmean(v) @ Wv^T.
// ---------------------------------------------------------------------------
__global__ __launch_bounds__(128)
void hc_state_kernel(const float* __restrict__ k, const float* __restrict__ v,
                     const float* __restrict__ Wk, const float* __restrict__ Wv,
                     float* __restrict__ states)
{
    const int t    = blockIdx.x;
    const int bh   = blockIdx.y;
    const int tid  = threadIdx.x;
    const int lane = tid & 31;
    const int w    = tid >> 5;      // wave id 0..3
    const int half = lane >> 4;
    const int lm   = lane & 15;

    __shared__ float kraw[CQ * DQ];     // packed, row-major [c][d]
    __shared__ float vraw[CQ * DQ];
    __shared__ float kmean[CQ * PS];    // padded
    __shared__ float vmean[CQ * PS];
    __shared__ float kp[CQ * PS];
    __shared__ float vp[CQ * PS];
    __shared__ float wkl[DQ * PS];      // Wk[level] staged, [e][d] padded
    __shared__ float wvl[DQ * PS];

    const size_t base = ((size_t)bh * NQ + (size_t)t * CQ) * DQ;
    {
        const float4* k4 = reinterpret_cast<const float4*>(k + base);
        const float4* v4 = reinterpret_cast<const float4*>(v + base);
        for (int i = tid; i < CQ * DQ / 4; i += 128) {
            reinterpret_cast<float4*>(kraw)[i] = k4[i];
            reinterpret_cast<float4*>(vraw)[i] = v4[i];
        }
    }

    for (int l = 0; l < LQ; ++l) {
        __syncthreads();
        // Stage projection weights for this level.
        for (int i = tid; i < DQ * DQ; i += 128) {
            wkl[(i >> 6) * PS + (i & 63)] = Wk[(size_t)l * DQ * DQ + i];
            wvl[(i >> 6) * PS + (i & 63)] = Wv[(size_t)l * DQ * DQ + i];
        }
        // Causal block means (block = 2^(l+1), <=16, aligned within chunk).
        const int blk = 2 << l;
        for (int i = tid; i < CQ * DQ; i += 128) {
            const int c = i >> 6, d = i & 63;
            const int start = c & ~(blk - 1);
            float sk = 0.0f, sv = 0.0f;
            for (int j = start; j <= c; ++j) {
                sk += kraw[j * DQ + d];
                sv += vraw[j * DQ + d];
            }
            const float inv = 1.0f / (float)(c - start + 1);
            kmean[c * PS + d] = sk * inv;
            vmean[c * PS + d] = sv * inv;
        }
        __syncthreads();

        // Projection: kp = elu1(kmean @ Wk^T), vp = vmean @ Wv^T. (WMMA)
        {
            const int row = w * 16 + lm;        // A row (per-lane)
            for (int et = 0; et < 4; ++et) {
                const int e0 = et * 16;
                v8f ak = {};
                v8f av = {};
                for (int k0 = 0; k0 < DQ; k0 += 4) {
                    const int ka = k0 + 2 * half;
                    ak = wmma4(ak, kmean[row * PS + ka], kmean[row * PS + ka + 1],
                               wkl[(e0 + lm) * PS + ka], wkl[(e0 + lm) * PS + ka + 1]);
                    av = wmma4(av, vmean[row * PS + ka], vmean[row * PS + ka + 1],
                               wvl[(e0 + lm) * PS + ka], wvl[(e0 + lm) * PS + ka + 1]);
                }
                for (int r = 0; r < 8; ++r) {
                    const int rr = w * 16 + r + 8 * half;
                    kp[rr * PS + e0 + lm] = elu1(ak[r]);
                    vp[rr * PS + e0 + lm] = av[r];
                }
            }
        }
        __syncthreads();

        // KV = kp^T @ vp  (A[m=d][kk=c] = kp[c][d]); Ksum.
        float* st = states + (((size_t)bh * LQ + l) * NCQ + t) * STATE_STRIDE;
        {
            const int drow = w * 16 + lm;
            for (int et = 0; et < 4; ++et) {
                const int e0 = et * 16;
                v8f acc = {};
                for (int k0 = 0; k0 < CQ; k0 += 4) {
                    const int kc = k0 + 2 * half;
                    acc = wmma4(acc, kp[kc * PS + drow], kp[(kc + 1) * PS + drow],
                                vp[kc * PS + e0 + lm], vp[(kc + 1) * PS + e0 + lm]);
                }
                for (int r = 0; r < 8; ++r) {
                    const int dd = w * 16 + r + 8 * half;
                    st[dd * DQ + e0 + lm] = acc[r];
                }
            }
        }
        if (tid < DQ) {
            float s = 0.0f;
            for (int c = 0; c < CQ; ++c) s += kp[c * PS + tid];
            st[DQ * DQ + tid] = s;
        }
    }
}

// ---------------------------------------------------------------------------
// Kernel 2: inclusive prefix scan of chunk states over the chunk axis.
// One block per (bh, level); 63 serial steps of 4160-float vector adds.
// ---------------------------------------------------------------------------
__global__ __launch_bounds__(256)
void hc_scan_kernel(float* __restrict__ states)
{
    float* base = states + (size_t)blockIdx.x * NCQ * STATE_STRIDE;
    for (int t = 1; t < NCQ; ++t) {
        float* cur        = base + (size_t)t * STATE_STRIDE;
        const float* prev = base + (size_t)(t - 1) * STATE_STRIDE;
        for (int i = threadIdx.x; i < STATE_STRIDE; i += 256) cur[i] += prev[i];
        __syncthreads();
    }
}

// ---------------------------------------------------------------------------
// Kernel 3: per-(bh, chunk), loop levels: recompute kp/vp, then
//   S = causal_mask(qp @ kp^T);  out = (S @ vp + qp @ KV_prev) / norm
//   norm = rowsum(S) + qp . Ksum_prev;  accumulate softmax(sw)[l]*out.
// ---------------------------------------------------------------------------
__global__ __launch_bounds__(128)
void hc_out_kernel(const float* __restrict__ q, const float* __restrict__ k,
                   const float* __restrict__ v, const float* __restrict__ Wk,
                   const float* __restrict__ Wv, const float* __restrict__ scale_w,
                   const float* __restrict__ states, float* __restrict__ out)
{
    const int t    = blockIdx.x;
    const int bh   = blockIdx.y;
    const int tid  = threadIdx.x;
    const int lane = tid & 31;
    const int w    = tid >> 5;
    const int half = lane >> 4;
    const int lm   = lane & 15;

    __shared__ float qp[CQ * PS];       // elu(q)+1, padded
    __shared__ float kraw[CQ * DQ];
    __shared__ float vraw[CQ * DQ];
    __shared__ float kp[CQ * PS];
    __shared__ float vp[CQ * PS];
    __shared__ float wkl[DQ * PS];
    __shared__ float wvl[DQ * PS];
    // scratch: phase A = kmean(CQ*PS) + vmean(CQ*PS); phase B = S(CQ*PS) + KVprev(DQ*DQ)
    __shared__ float scratch[2 * CQ * PS];
    __shared__ float ksum_prev[DQ];
    __shared__ float norms[CQ];

    float* kmean = scratch;
    float* vmean = scratch + CQ * PS;
    float* Sm    = scratch;             // padded stride PS, rows=chunk pos
    float* KVp   = scratch + CQ * PS;   // packed [d][e]

    // Softmax of scale weights (tiny, redundant per thread).
    const float s0 = scale_w[0], s1 = scale_w[1], s2 = scale_w[2], s3 = scale_w[3];
    const float mx = fmaxf(fmaxf(s0, s1), fmaxf(s2, s3));
    const float wsum = __expf(s0 - mx) + __expf(s1 - mx) + __expf(s2 - mx) + __expf(s3 - mx);

    const size_t base = ((size_t)bh * NQ + (size_t)t * CQ) * DQ;
    for (int i = tid; i < CQ * DQ; i += 128) {
        const int c = i >> 6, d = i & 63;
        qp[c * PS + d] = elu1(q[base + i]);
        kraw[i] = k[base + i];
        vraw[i] = v[base + i];
    }

    v8f oacc[4];
    for (int et = 0; et < 4; ++et) oacc[et] = {};

    for (int l = 0; l < LQ; ++l) {
        const float swl = __expf(scale_w[l] - mx) / wsum;
        __syncthreads();   // previous level fully consumed

        for (int i = tid; i < DQ * DQ; i += 128) {
            wkl[(i >> 6) * PS + (i & 63)] = Wk[(size_t)l * DQ * DQ + i];
            wvl[(i >> 6) * PS + (i & 63)] = Wv[(size_t)l * DQ * DQ + i];
        }
        const int blk = 2 << l;
        for (int i = tid; i < CQ * DQ; i += 128) {
            const int c = i >> 6, d = i & 63;
            const int start = c & ~(blk - 1);
            float sk = 0.0f, sv = 0.0f;
            for (int j = start; j <= c; ++j) {
                sk += kraw[j * DQ + d];
                sv += vraw[j * DQ + d];
            }
            const float inv = 1.0f / (float)(c - start + 1);
            kmean[c * PS + d] = sk * inv;
            vmean[c * PS + d] = sv * inv;
        }
        __syncthreads();

        // Projection into kp / vp (WMMA), elu+1 on k side.
        {
            const int row = w * 16 + lm;
            for (int et = 0; et < 4; ++et) {
                const int e0 = et * 16;
                v8f ak = {};
                v8f av = {};
                for (int k0 = 0; k0 < DQ; k0 += 4) {
                    const int ka = k0 + 2 * half;
                    ak = wmma4(ak, kmean[row * PS + ka], kmean[row * PS + ka + 1],
                               wkl[(e0 + lm) * PS + ka], wkl[(e0 + lm) * PS + ka + 1]);
                    av = wmma4(av, vmean[row * PS + ka], vmean[row * PS + ka + 1],
                               wvl[(e0 + lm) * PS + ka], wvl[(e0 + lm) * PS + ka + 1]);
                }
                for (int r = 0; r < 8; ++r) {
                    const int rr = w * 16 + r + 8 * half;
                    kp[rr * PS + e0 + lm] = elu1(ak[r]);
                    vp[rr * PS + e0 + lm] = av[r];
                }
            }
        }
        __syncthreads();   // means consumed -> scratch reusable

        // Load inter-chunk carried state (exclusive prefix = inclusive[t-1]).
        const float* st = states +
            (((size_t)bh * LQ + l) * NCQ + (size_t)(t > 0 ? t - 1 : 0)) * STATE_STRIDE;
        const float zf = (t > 0) ? 1.0f : 0.0f;
        for (int i = tid; i < DQ * DQ; i += 128) KVp[i] = zf * st[i];
        if (tid < DQ) ksum_prev[tid] = zf * st[DQ * DQ + tid];
        __syncthreads();

        // S strip (16 rows per wave): tiles jt=0..w (causal), masked diag tile.
        for (int jt = 0; jt <= w; ++jt) {
            const int j0 = jt * 16;
            v8f acc = {};
            for (int k0 = 0; k0 < DQ; k0 += 4) {
                const int kk = k0 + 2 * half;
                acc = wmma4(acc, qp[(w * 16 + lm) * PS + kk], qp[(w * 16 + lm) * PS + kk + 1],
                            kp[(j0 + lm) * PS + kk], kp[(j0 + lm) * PS + kk + 1]);
            }
            for (int r = 0; r < 8; ++r) {
                const int row = w * 16 + r + 8 * half;
                const int col = j0 + lm;
                Sm[row * PS + col] = (col <= row) ? acc[r] : 0.0f;
            }
        }
        __syncthreads();

        // Per-row normalizer.
        if (tid < CQ) {
            const int row = tid;
            float s = 0.0f;
            for (int j = 0; j <= row; ++j) s += Sm[row * PS + j];
            for (int d = 0; d < DQ; ++d) s += qp[row * PS + d] * ksum_prev[d];
            norms[row] = fmaxf(s, EPSQ);
        }
        __syncthreads();

        // out_strip = S @ vp + qp @ KV_prev (WMMA), then normalize + weight.
        {
            const int kmaxS = (w + 1) * 16;   // cols beyond strip are all zero
            for (int et = 0; et < 4; ++et) {
                const int e0 = et * 16;
                v8f acc = {};
                for (int k0 = 0; k0 < kmaxS; k0 += 4) {
                    const int kk = k0 + 2 * half;
                    acc = wmma4(acc, Sm[(w * 16 + lm) * PS + kk], Sm[(w * 16 + lm) * PS + kk + 1],
                                vp[kk * PS + e0 + lm], vp[(kk + 1) * PS + e0 + lm]);
                }
                for (int k0 = 0; k0 < DQ; k0 += 4) {
                    const int kk = k0 + 2 * half;
                    acc = wmma4(acc, qp[(w * 16 + lm) * PS + kk], qp[(w * 16 + lm) * PS + kk + 1],
                                KVp[kk * DQ + e0 + lm], KVp[(kk + 1) * DQ + e0 + lm]);
                }
                for (int r = 0; r < 8; ++r) {
                    const int row = w * 16 + r + 8 * half;
                    oacc[et][r] += swl * acc[r] / norms[row];
                }
            }
        }
    }

    // Final store (each block owns its 64x64 output slab; no atomics).
    for (int et = 0; et < 4; ++et) {
        for (int r = 0; r < 8; ++r) {
            const int row = w * 16 + r + 8 * half;
            out[base + (size_t)row * DQ + et * 16 + lm] = oacc[et][r];
        }
    }
}

// ---------------------------------------------------------------------------
// Launch. d_ws must hold BH*L*NC*4160 floats = ~130 MB of chunk states.
// ---------------------------------------------------------------------------
extern "C" void kernel_launch(void* const* d_in, const int* in_sizes, int n_in,
                              void* d_out, int out_size, void* d_ws, size_t ws_size,
                              hipStream_t stream) {
    (void)in_sizes; (void)n_in; (void)out_size; (void)ws_size;
    const float* q  = (const float*)d_in[0];
    const float* k  = (const float*)d_in[1];
    const float* v  = (const float*)d_in[2];
    const float* Wk = (const float*)d_in[3];
    const float* Wv = (const float*)d_in[4];
    const float* sw = (const float*)d_in[5];
    // d_in[6] = chunk_size: ignored (our internal chunking is math-equivalent).
    float* out    = (float*)d_out;
    float* states = (float*)d_ws;

    dim3 grid(NCQ, BHQ);
    hipLaunchKernelGGL(hc_state_kernel, grid, dim3(128), 0, stream, k, v, Wk, Wv, states);
    hipLaunchKernelGGL(hc_scan_kernel, dim3(BHQ * LQ), dim3(256), 0, stream, states);
    hipLaunchKernelGGL(hc_out_kernel, grid, dim3(128), 0, stream,
                       q, k, v, Wk, Wv, sw, states, out);
}